// SS2D_pure_6021544149044
// MI455X (gfx1250) — compile-verified
//
#include <hip/hip_runtime.h>
#include <math.h>

// ---------------- problem constants ----------------
constexpr int BB = 2, HH = 64, WW = 64, DD = 192;
constexpr int NS = 16, RR = 12, KK = 4;
constexpr int LL = HH * WW;              // 4096
constexpr int NC = 16, CL = LL / NC;     // 16 chunks of 256 steps
constexpr int COUT = RR + 2 * NS;        // 44

typedef __attribute__((ext_vector_type(2))) float v2f;
typedef __attribute__((ext_vector_type(8))) float v8f;

// ---------------- workspace layout (floats) ----------------
constexpr size_t OFF_BC = 0;                           // (B*K, L, 32)  B||C per step
constexpr size_t SZ_BC  = (size_t)BB * KK * LL * 32;
constexpr size_t OFF_DT = OFF_BC + SZ_BC;              // (B*K, L, D)   softplus(dt)
constexpr size_t SZ_DT  = (size_t)BB * KK * LL * DD;
constexpr size_t SZ_SUM = (size_t)BB * KK * NC * DD * NS;
constexpr size_t OFF_AP = OFF_DT + SZ_DT;              // chunk prod(dA)
constexpr size_t OFF_HE = OFF_AP + SZ_SUM;             // chunk h_end (h_in=0)
constexpr size_t OFF_HI = OFF_HE + SZ_SUM;             // chunk h_in (actual)
constexpr size_t OFF_Y4 = OFF_HI + SZ_SUM;             // (B, L(p), D, K) per-dir y

// direction-k scan index l  ->  row-major spatial position p = h*W + w
__device__ __forceinline__ int dir_pos(int k, int l) {
    if (k == 0) return l;
    if (k == 1) { return (l & 63) * WW + (l >> 6); }
    if (k == 2) return (LL - 1) - l;
    int lr = (LL - 1) - l;
    return (lr & 63) * WW + (lr >> 6);
}

// Async-copy one chunk's B||C tile (CL rows x 32 floats = 32 KB) into LDS.
// Uses gfx1250 GLOBAL_LOAD_ASYNC_TO_LDS_B128 (ASYNCcnt) + s_wait_asynccnt.
__device__ __forceinline__ void stage_bc_async(const float* __restrict__ src,
                                               float* __restrict__ sbc)
{
    const uint32_t lds_base = (uint32_t)(size_t)sbc;   // low 32 bits = LDS offset
    for (int i = threadIdx.x; i < CL * 32 / 4; i += blockDim.x) {
        uint32_t loff = lds_base + (uint32_t)i * 16u;
        uint64_t gaddr = (uint64_t)(size_t)(src + (size_t)i * 4);
        asm volatile("global_load_async_to_lds_b128 %0, %1, off"
                     :: "v"(loff), "v"(gaddr) : "memory");
    }
    asm volatile("s_wait_asynccnt 0x0" ::: "memory");
    __syncthreads();
}

// =====================================================================
// Kernel 1: fused cross-scan gather + x_proj GEMM (44x192) +
//           dt_proj GEMM (192x12) + bias + softplus.   f32 WMMA 16x16x4.
// grid = (L/16/4, B*K), block = 128 (4 waves, 1 l-tile per wave)
// =====================================================================
__global__ __launch_bounds__(128) void ss2d_proj(
    const float* __restrict__ x, const float* __restrict__ wproj,
    const float* __restrict__ wdt, const float* __restrict__ bdt,
    float* __restrict__ ws)
{
    float* BC = ws + OFF_BC;
    float* DT = ws + OFF_DT;
    const int bk = blockIdx.y;
    const int b = bk >> 2, k = bk & 3;
    const int lane = threadIdx.x & 31, wave = threadIdx.x >> 5;
    const int l0 = (blockIdx.x * 4 + wave) * 16;
    const int n = lane & 15;          // A-row m and B/D-col n share this value
    const int khalf = lane >> 4;

    __shared__ float sdt[4][12][17];  // per-wave dt_rank tile (row-major, padded)

    // gather pointer for this lane's column l = l0 + n
    const int l = l0 + n;
    const int p = dir_pos(k, l);
    const float* xrow = x + (size_t)(b * LL + p) * DD;
    const float* w1 = wproj + (size_t)k * COUT * DD;

    v8f acc0 = {}, acc1 = {}, acc2 = {};
    for (int d0 = 0; d0 < DD; d0 += 4) {
        // B fragment: xs[d0+kk, l0+n], kk = 2*khalf + {0,1}  (contiguous in D)
        v2f bf = *(const v2f*)(xrow + d0 + 2 * khalf);
        // A fragments: W[c0+m, d0+kk]
        v2f a0 = *(const v2f*)(w1 + (size_t)(n) * DD + d0 + 2 * khalf);
        acc0 = __builtin_amdgcn_wmma_f32_16x16x4_f32(false, a0, false, bf,
                                                     (short)0, acc0, false, false);
        v2f a1 = *(const v2f*)(w1 + (size_t)(16 + n) * DD + d0 + 2 * khalf);
        acc1 = __builtin_amdgcn_wmma_f32_16x16x4_f32(false, a1, false, bf,
                                                     (short)0, acc1, false, false);
        int c2 = 32 + n;
        float msk = (c2 < COUT) ? 1.f : 0.f;
        v2f a2 = *(const v2f*)(w1 + (size_t)(c2 < COUT ? c2 : COUT - 1) * DD + d0 + 2 * khalf);
        a2.x *= msk; a2.y *= msk;
        acc2 = __builtin_amdgcn_wmma_f32_16x16x4_f32(false, a2, false, bf,
                                                     (short)0, acc2, false, false);
    }

    // scatter D-tiles: rows 0..11 -> LDS (dt_rank), 12..27 -> B, 28..43 -> C
    #pragma unroll
    for (int mt = 0; mt < 3; ++mt) {
        v8f a = (mt == 0) ? acc0 : ((mt == 1) ? acc1 : acc2);
        #pragma unroll
        for (int r = 0; r < 8; ++r) {
            int c = mt * 16 + r + 8 * khalf;
            float v = a[r];
            size_t row = (size_t)(bk * LL + l) * 32;
            if (c < 12)       sdt[wave][c][n] = v;
            else if (c < 28)  BC[row + (c - 12)] = v;
            else if (c < 44)  BC[row + 16 + (c - 28)] = v;
        }
    }
    __syncthreads();

    // GEMM2: dt[d, l] = softplus( sum_r Wd[d,r] * sdt[r,l] + bias[d] )
    const float* w2 = wdt + (size_t)k * DD * RR;
    for (int mt = 0; mt < 12; ++mt) {
        const int dbase = mt * 16;
        v8f acc = {};
        #pragma unroll
        for (int r0 = 0; r0 < RR; r0 += 4) {
            v2f bf;
            bf.x = sdt[wave][r0 + 2 * khalf][n];
            bf.y = sdt[wave][r0 + 2 * khalf + 1][n];
            v2f af = *(const v2f*)(w2 + (size_t)(dbase + n) * RR + r0 + 2 * khalf);
            acc = __builtin_amdgcn_wmma_f32_16x16x4_f32(false, af, false, bf,
                                                        (short)0, acc, false, false);
        }
        #pragma unroll
        for (int r = 0; r < 8; ++r) {
            int d = dbase + r + 8 * khalf;
            float v = acc[r] + bdt[k * DD + d];
            v = (v > 20.f) ? v : log1pf(expf(v));     // softplus
            DT[(size_t)(bk * LL + l) * DD + d] = v;
        }
    }
}

// =====================================================================
// Kernel 2: per-chunk scan summaries: prod(dA) and h_end with h_in = 0
// grid = B*K*NC, block = 192. B||C tile staged to LDS via async copy.
// =====================================================================
__global__ __launch_bounds__(192) void ss2d_scan_sum(
    const float* __restrict__ x, const float* __restrict__ A_logs,
    float* __restrict__ ws)
{
    const float* DT = ws + OFF_DT;
    const float* BC = ws + OFF_BC;
    float* AP = ws + OFF_AP;
    float* HE = ws + OFF_HE;
    const int bk = blockIdx.x / NC, c = blockIdx.x % NC;
    const int b = bk >> 2, k = bk & 3;
    const int d = threadIdx.x;
    const int l0 = c * CL;

    __shared__ __align__(16) float sbc[CL * 32];      // 32 KB chunk tile
    stage_bc_async(BC + (size_t)(bk * LL + l0) * 32, sbc);

    float Aa[NS], h[NS], ap[NS];
    #pragma unroll
    for (int nn = 0; nn < NS; ++nn) {
        Aa[nn] = -expf(A_logs[(size_t)(k * DD + d) * NS + nn]);
        h[nn] = 0.f; ap[nn] = 1.f;
    }
    for (int t = 0; t < CL; ++t) {
        const int l = l0 + t;
        const int p = dir_pos(k, l);
        float xv  = x[(size_t)(b * LL + p) * DD + d];
        float dtv = DT[(size_t)(bk * LL + l) * DD + d];
        const float* bcrow = &sbc[t * 32];            // LDS broadcast
        float dx = dtv * xv;
        #pragma unroll
        for (int nn = 0; nn < NS; ++nn) {
            float dA = expf(dtv * Aa[nn]);
            h[nn] = dA * h[nn] + dx * bcrow[nn];
            ap[nn] *= dA;
        }
    }
    size_t base = ((size_t)(bk * NC + c) * DD + d) * NS;
    #pragma unroll
    for (int nn = 0; nn < NS; ++nn) { HE[base + nn] = h[nn]; AP[base + nn] = ap[nn]; }
}

// =====================================================================
// Kernel 3: tiny serial prefix over the 16 chunks -> actual h_in per chunk
// =====================================================================
__global__ __launch_bounds__(256) void ss2d_chunk_prefix(float* __restrict__ ws)
{
    const int idx = blockIdx.x * 256 + threadIdx.x;
    if (idx >= BB * KK * DD) return;
    const int bk = idx / DD, d = idx % DD;
    const float* AP = ws + OFF_AP;
    const float* HE = ws + OFF_HE;
    float* HI = ws + OFF_HI;
    float h[NS];
    #pragma unroll
    for (int nn = 0; nn < NS; ++nn) h[nn] = 0.f;
    for (int c = 0; c < NC; ++c) {
        size_t base = ((size_t)(bk * NC + c) * DD + d) * NS;
        #pragma unroll
        for (int nn = 0; nn < NS; ++nn) {
            HI[base + nn] = h[nn];
            h[nn] = HE[base + nn] + AP[base + nn] * h[nn];
        }
    }
}

// =====================================================================
// Kernel 4: full scan with correct h_in; y = <h,C> + Ds*x, written at the
// merged spatial position p so cross_merge becomes a simple 4-way sum.
// =====================================================================
__global__ __launch_bounds__(192) void ss2d_scan_out(
    const float* __restrict__ x, const float* __restrict__ A_logs,
    const float* __restrict__ Ds, float* __restrict__ ws)
{
    const float* DT = ws + OFF_DT;
    const float* BC = ws + OFF_BC;
    const float* HI = ws + OFF_HI;
    float* Y4 = ws + OFF_Y4;
    const int bk = blockIdx.x / NC, c = blockIdx.x % NC;
    const int b = bk >> 2, k = bk & 3;
    const int d = threadIdx.x;
    const int l0 = c * CL;

    __shared__ __align__(16) float sbc[CL * 32];      // 32 KB chunk tile
    stage_bc_async(BC + (size_t)(bk * LL + l0) * 32, sbc);

    float Aa[NS], h[NS];
    size_t hbase = ((size_t)(bk * NC + c) * DD + d) * NS;
    #pragma unroll
    for (int nn = 0; nn < NS; ++nn) {
        Aa[nn] = -expf(A_logs[(size_t)(k * DD + d) * NS + nn]);
        h[nn] = HI[hbase + nn];
    }
    const float dskip = Ds[k * DD + d];
    for (int t = 0; t < CL; ++t) {
        const int l = l0 + t;
        const int p = dir_pos(k, l);
        float xv  = x[(size_t)(b * LL + p) * DD + d];
        float dtv = DT[(size_t)(bk * LL + l) * DD + d];
        const float* bcrow = &sbc[t * 32];            // LDS broadcast
        float dx = dtv * xv;
        float y = 0.f;
        #pragma unroll
        for (int nn = 0; nn < NS; ++nn) {
            float dA = expf(dtv * Aa[nn]);
            h[nn] = dA * h[nn] + dx * bcrow[nn];
            y += h[nn] * bcrow[16 + nn];
        }
        y += dskip * xv;
        Y4[((size_t)(b * LL + p) * DD + d) * KK + k] = y;
    }
}

// =====================================================================
// Kernel 5: 4-way merge + LayerNorm over D, write (B,H,W,D) output
// grid = B*L, block = 192
// =====================================================================
__global__ __launch_bounds__(192) void ss2d_merge_ln(
    const float* __restrict__ ln_w, const float* __restrict__ ln_b,
    const float* __restrict__ ws, float* __restrict__ out)
{
    const float* Y4 = ws + OFF_Y4;
    const size_t bp = blockIdx.x;
    const int d = threadIdx.x;
    const float4 v4 = *(const float4*)(Y4 + (bp * DD + d) * 4);
    const float y = v4.x + v4.y + v4.z + v4.w;

    __shared__ float sred[6];
    const int lane = threadIdx.x & 31, wv = threadIdx.x >> 5;

    float v = y;
    #pragma unroll
    for (int off = 16; off > 0; off >>= 1) v += __shfl_down(v, off, 32);
    if (lane == 0) sred[wv] = v;
    __syncthreads();
    float s = 0.f;
    #pragma unroll
    for (int i = 0; i < 6; ++i) s += sred[i];
    const float mean = s / (float)DD;
    const float dy = y - mean;
    __syncthreads();

    v = dy * dy;
    #pragma unroll
    for (int off = 16; off > 0; off >>= 1) v += __shfl_down(v, off, 32);
    if (lane == 0) sred[wv] = v;
    __syncthreads();
    s = 0.f;
    #pragma unroll
    for (int i = 0; i < 6; ++i) s += sred[i];
    const float var = s / (float)DD;

    out[bp * DD + d] = dy * rsqrtf(var + 1e-5f) * ln_w[d] + ln_b[d];
}

// =====================================================================
extern "C" void kernel_launch(void* const* d_in, const int* in_sizes, int n_in,
                              void* d_out, int out_size, void* d_ws, size_t ws_size,
                              hipStream_t stream)
{
    const float* x    = (const float*)d_in[0];
    const float* wp   = (const float*)d_in[1];
    const float* wdt  = (const float*)d_in[2];
    const float* bdt  = (const float*)d_in[3];
    const float* alog = (const float*)d_in[4];
    const float* Ds   = (const float*)d_in[5];
    const float* lw   = (const float*)d_in[6];
    const float* lb   = (const float*)d_in[7];
    float* ws  = (float*)d_ws;
    float* out = (float*)d_out;

    ss2d_proj<<<dim3(LL / 16 / 4, BB * KK), 128, 0, stream>>>(x, wp, wdt, bdt, ws);
    ss2d_scan_sum<<<BB * KK * NC, DD, 0, stream>>>(x, alog, ws);
    ss2d_chunk_prefix<<<(BB * KK * DD + 255) / 256, 256, 0, stream>>>(ws);
    ss2d_scan_out<<<BB * KK * NC, DD, 0, stream>>>(x, alog, Ds, ws);
    ss2d_merge_ln<<<BB * LL, DD, 0, stream>>>(lw, lb, ws, out);
}